// GraphSAGE_32220844655372
// MI455X (gfx1250) — compile-verified
//
#include <hip/hip_runtime.h>
#include <hip/hip_bf16.h>

typedef float v2f __attribute__((ext_vector_type(2)));
typedef float v8f __attribute__((ext_vector_type(8)));

#define FEAT 128          // D == H == 128 for every layer input
#define RT_PER_BLOCK 16   // row tiles per block (weight reuse factor)
#define SA_STRIDE 132     // padded A-row stride in LDS (kills 128%64==0 conflicts)

// ---------------------------------------------------------------------------
// utility kernels
// ---------------------------------------------------------------------------
__global__ void sage_zero_f32(float* __restrict__ p, long n) {
    long i = (long)blockIdx.x * blockDim.x + threadIdx.x;
    long stride = (long)gridDim.x * blockDim.x;
    for (; i < n; i += stride) p[i] = 0.0f;
}

__global__ void sage_degree(const int* __restrict__ dst, float* __restrict__ deg, int E) {
    int e = blockIdx.x * blockDim.x + threadIdx.x;
    if (e < E) atomicAdd(&deg[dst[e]], 1.0f);
}

__global__ void sage_inv_deg(float* __restrict__ deg, int n) {
    int i = blockIdx.x * blockDim.x + threadIdx.x;
    if (i < n) deg[i] = 1.0f / fmaxf(deg[i], 1.0f);
}

// scatter-add, float4-vectorized: 32 threads per edge, 4 floats per thread
__global__ void sage_scatter4(const float* __restrict__ h,
                              const int* __restrict__ src,
                              const int* __restrict__ dst,
                              float* __restrict__ agg, int E) {
    long idx = (long)blockIdx.x * blockDim.x + threadIdx.x;
    long total = (long)E * 32;
    if (idx >= total) return;
    int e = (int)(idx >> 5);
    int q = (int)(idx & 31) << 2;
    int s = src[e];
    int d = dst[e];
    const float4 v = *(const float4*)(h + (long)s * FEAT + q);
    float* p = agg + (long)d * FEAT + q;
    atomicAdd(p + 0, v.x);
    atomicAdd(p + 1, v.y);
    atomicAdd(p + 2, v.z);
    atomicAdd(p + 3, v.w);
}

// in-place mean, float4-vectorized
__global__ void sage_normalize4(float* __restrict__ agg,
                                const float* __restrict__ inv_deg, int n_nodes) {
    long idx = (long)blockIdx.x * blockDim.x + threadIdx.x;
    long total = (long)n_nodes * 32;
    if (idx >= total) return;
    int node = (int)(idx >> 5);
    int q = (int)(idx & 31) << 2;
    float w = inv_deg[node];
    float4* p = (float4*)(agg + (long)node * FEAT + q);
    float4 v = *p;
    v.x *= w; v.y *= w; v.z *= w; v.w *= w;
    *p = v;
}

// ---------------------------------------------------------------------------
// LDS-staged fused SAGE GEMM: out = act( hin @ Ws + hneigh @ Wn + b )
// Weight matrices live in LDS split into 64-K-row halves (<64KB each) so the
// fully unrolled K-loop addresses everything as loop-invariant-base +
// 16-bit immediate ds offsets. Each wave computes a 16x16 tile with
// V_WMMA_F32_16X16X4_F32.
// ---------------------------------------------------------------------------
template <int NC>
__global__ __launch_bounds__(256)
void sage_gemm_wmma_lds(const float* __restrict__ hin,
                        const float* __restrict__ hneigh,
                        const float* __restrict__ Ws,
                        const float* __restrict__ Wn,
                        const float* __restrict__ bias,
                        float* __restrict__ out,
                        int n_rows, int do_relu) {
    constexpr int WSTRIDE = NC + 8;            // padded weight-row stride
    constexpr int WHALF   = 64 * WSTRIDE;      // one 64-K-row half (<16K floats)
    extern __shared__ float smem[];
    // layout: [selfK0-63][selfK64-127][neighK0-63][neighK64-127][A self][A neigh]
    float* sW = smem;
    float* sA = smem + 4 * WHALF;

    const int tid  = threadIdx.x;
    const int wave = tid >> 5;
    const int lane = tid & 31;
    constexpr int col_tiles = (NC + 15) >> 4;

    // ---- stage weights once per block (row k of W -> half k/64, row k%64) ----
    for (int i = tid; i < FEAT * NC; i += 256) {
        int k = i / NC;
        int c = i - k * NC;
        int half = k >> 6;
        int kk = k & 63;
        sW[half * WHALF + kk * WSTRIDE + c]            = Ws[i];
        sW[(2 + half) * WHALF + kk * WSTRIDE + c]      = Wn[i];
    }

    // ---- per-lane fragment geometry ----
    const int m     = lane & 15;               // A row within tile
    const int khalf = (lane >> 4) << 1;        // 0 or 2 (K split across half-waves)
    const int ct    = (wave < col_tiles) ? wave : 0;
    const int col   = (ct << 4) + (lane & 15);
    const bool cv   = col < NC;
    const int ccol  = cv ? col : (NC - 1);

    // six loop-invariant LDS bases; all inner-loop offsets are immediates
    const float* __restrict__ bS0 = sW + 0 * WHALF + khalf * WSTRIDE + ccol;
    const float* __restrict__ bS1 = sW + 1 * WHALF + khalf * WSTRIDE + ccol;
    const float* __restrict__ bN0 = sW + 2 * WHALF + khalf * WSTRIDE + ccol;
    const float* __restrict__ bN1 = sW + 3 * WHALF + khalf * WSTRIDE + ccol;
    const float* __restrict__ aSb = sA + m * SA_STRIDE + khalf;
    const float* __restrict__ aNb = sA + 16 * SA_STRIDE + m * SA_STRIDE + khalf;

    const int rt0 = blockIdx.x * RT_PER_BLOCK;
    for (int rti = 0; rti < RT_PER_BLOCK; ++rti) {
        const int row0 = (rt0 + rti) << 4;
        if (row0 >= n_rows) break;             // block-uniform

        // ---- stage 16x128 A rows (self + neigh), float4, padded stride ----
        __syncthreads();                       // prior iteration done reading sA
        for (int i = tid; i < 16 * (FEAT / 4); i += 256) {   // 512 float4 slots
            int r = i >> 5;
            int f = (i & 31) << 2;
            const float4 vs = *(const float4*)&hin   [(long)(row0 + r) * FEAT + f];
            const float4 vn = *(const float4*)&hneigh[(long)(row0 + r) * FEAT + f];
            *(float4*)&sA[r * SA_STRIDE + f]                  = vs;
            *(float4*)&sA[16 * SA_STRIDE + r * SA_STRIDE + f] = vn;
        }
        __syncthreads();                       // sA (and first-pass sW) visible

        if (wave < col_tiles) {                // wave-uniform: EXEC all-1 at WMMA
            v8f acc = {0.f, 0.f, 0.f, 0.f, 0.f, 0.f, 0.f, 0.f};

            auto step = [&](const float* __restrict__ bSr,
                            const float* __restrict__ bNr, int kA, int kB) {
                v2f a, b;
                a.x = aSb[kA];
                a.y = aSb[kA + 1];
                b.x = bSr[kB * WSTRIDE];
                b.y = bSr[(kB + 1) * WSTRIDE];
                acc = __builtin_amdgcn_wmma_f32_16x16x4_f32(false, a, false, b,
                                                            (short)0, acc, false, false);
                a.x = aNb[kA];
                a.y = aNb[kA + 1];
                b.x = bNr[kB * WSTRIDE];
                b.y = bNr[(kB + 1) * WSTRIDE];
                acc = __builtin_amdgcn_wmma_f32_16x16x4_f32(false, a, false, b,
                                                            (short)0, acc, false, false);
            };

            #pragma unroll
            for (int k0 = 0; k0 < 64; k0 += 4)   // K = 0..63
                step(bS0, bN0, k0, k0);
            #pragma unroll
            for (int k0 = 0; k0 < 64; k0 += 4)   // K = 64..127
                step(bS1, bN1, k0 + 64, k0);

            // D layout: VGPR v -> M = v + (lane>=16 ? 8 : 0), N = lane&15
            const float bv = bias[ccol];
            const int mbase = (lane >> 4) << 3;
            #pragma unroll
            for (int v = 0; v < 8; ++v) {
                const int r = row0 + mbase + v;
                if (cv && r < n_rows) {
                    float val = acc[v] + bv;
                    if (do_relu) val = fmaxf(val, 0.0f);
                    out[(long)r * NC + (long)ccol] = val;
                }
            }
        }
    }
}

// ---------------------------------------------------------------------------
// host-side orchestration
// ---------------------------------------------------------------------------
template <int NC>
static void launch_gemm(const float* hin, const float* hneigh,
                        const float* Ws, const float* Wn, const float* b,
                        float* out, int N, int relu, hipStream_t stream) {
    constexpr int WSTRIDE = NC + 8;
    const size_t shmem = (size_t)(4 * 64 * WSTRIDE + 2 * 16 * SA_STRIDE) * sizeof(float);
    int row_tiles = (N + 15) >> 4;
    int blocks = (row_tiles + RT_PER_BLOCK - 1) / RT_PER_BLOCK;
    sage_gemm_wmma_lds<NC><<<blocks, 256, shmem, stream>>>(hin, hneigh, Ws, Wn, b,
                                                           out, N, relu);
}

static void aggregate(const float* hin, float* agg, const float* inv_deg,
                      const int* src, const int* dst, int N, int E,
                      hipStream_t stream) {
    const long featN = (long)N * FEAT;
    sage_zero_f32<<<2048, 256, 0, stream>>>(agg, featN);
    {
        long total = (long)E * 32;
        int blocks = (int)((total + 255) / 256);
        sage_scatter4<<<blocks, 256, 0, stream>>>(hin, src, dst, agg, E);
    }
    {
        long total = (long)N * 32;
        int blocks = (int)((total + 255) / 256);
        sage_normalize4<<<blocks, 256, 0, stream>>>(agg, inv_deg, N);
    }
}

extern "C" void kernel_launch(void* const* d_in, const int* in_sizes, int n_in,
                              void* d_out, int out_size, void* d_ws, size_t ws_size,
                              hipStream_t stream) {
    // input order: x, src, dst, Ws0, Wn0, b0, Ws1, Wn1, b1, Ws2, Wn2, b2
    const float* x    = (const float*)d_in[0];
    const int*   src  = (const int*)d_in[1];
    const int*   dst  = (const int*)d_in[2];
    const float* Ws0  = (const float*)d_in[3];
    const float* Wn0  = (const float*)d_in[4];
    const float* b0   = (const float*)d_in[5];
    const float* Ws1  = (const float*)d_in[6];
    const float* Wn1  = (const float*)d_in[7];
    const float* b1   = (const float*)d_in[8];
    const float* Ws2  = (const float*)d_in[9];
    const float* Wn2  = (const float*)d_in[10];
    const float* b2   = (const float*)d_in[11];

    const int N = in_sizes[0] / FEAT;   // 50000
    const int E = in_sizes[1];          // 600000

    // workspace carve-out (floats)
    float* ws = (float*)d_ws;
    const long featN = (long)N * FEAT;
    float* agg     = ws;                    // N*128  (agg -> h_neigh in place)
    float* h1      = ws + featN;            // N*128
    float* h2      = ws + 2 * featN;        // N*128
    float* inv_deg = ws + 3 * featN;        // N

    // degree -> 1/max(deg,1)   (recomputed every call: deterministic)
    sage_zero_f32<<<256, 256, 0, stream>>>(inv_deg, (long)N);
    sage_degree<<<(E + 255) / 256, 256, 0, stream>>>(dst, inv_deg, E);
    sage_inv_deg<<<(N + 255) / 256, 256, 0, stream>>>(inv_deg, N);

    // layer 0: x -> h1 (relu)
    aggregate(x, agg, inv_deg, src, dst, N, E, stream);
    launch_gemm<128>(x,  agg, Ws0, Wn0, b0, h1, N, 1, stream);
    // layer 1: h1 -> h2 (relu)
    aggregate(h1, agg, inv_deg, src, dst, N, E, stream);
    launch_gemm<128>(h1, agg, Ws1, Wn1, b1, h2, N, 1, stream);
    // layer 2: h2 -> out (no relu, C = 10 cols)
    aggregate(h2, agg, inv_deg, src, dst, N, E, stream);
    launch_gemm<10>(h2, agg, Ws2, Wn2, b2, (float*)d_out, N, 0, stream);
}